// DilatedContextAttentionModule_75505525064628
// MI455X (gfx1250) — compile-verified
//
#include <hip/hip_runtime.h>
#include <cstddef>

typedef float v2f __attribute__((ext_vector_type(2)));
typedef float v8f __attribute__((ext_vector_type(8)));

static constexpr int Bb = 8;
static constexpr int Cc = 256;
static constexpr int Nn = 4096;           // H*W = 64*64
static constexpr float BN_EPS = 1e-5f;

// ---------------------------------------------------------------------------
// Kernel 1: M_b = Xj_b @ Xj_b^T   (C x C per batch, K = N = 4096), fp32 WMMA
// Block = 512 threads (16 waves). Each block computes a 64x64 tile of M_b.
// Each wave computes one 16x16 WMMA tile. LDS staging Kt=64, stride 68 floats
// (272 B: 16B-aligned rows, bank-rotated by 4 per row -> conflict-free frags).
// ---------------------------------------------------------------------------
__global__ __launch_bounds__(512) void k_syrk(const float* __restrict__ xj,
                                              float* __restrict__ M) {
    const int b  = blockIdx.z;
    const int bi = blockIdx.y;   // row 64-block (0..3)
    const int bj = blockIdx.x;   // col 64-block (0..3)
    __shared__ float lds_i[64 * 68];
    __shared__ float lds_j[64 * 68];
    const int tid  = threadIdx.x;
    const int wv   = tid >> 5;
    const int lane = tid & 31;
    const int half = lane >> 4;
    const int rl   = lane & 15;
    const int tr   = wv >> 2;    // wave's 16-row tile (0..3)
    const int tc   = wv & 3;     // wave's 16-col tile (0..3)
    const float* Xb = xj + (size_t)b * Cc * Nn;

    v8f acc = {};
    for (int kt = 0; kt < Nn; kt += 64) {
        // Stage two 64x64 fp32 operand blocks (rows i-block and j-block of Xj)
        for (int slot = tid; slot < 64 * 16; slot += 512) {
            const int row = slot >> 4;
            const int c4  = (slot & 15) << 2;
            const float4 vi = *(const float4*)(Xb + (size_t)(bi * 64 + row) * Nn + kt + c4);
            const float4 vj = *(const float4*)(Xb + (size_t)(bj * 64 + row) * Nn + kt + c4);
            *(float4*)(&lds_i[row * 68 + c4]) = vi;
            *(float4*)(&lds_j[row * 68 + c4]) = vj;
            if (kt + 64 < Nn) {  // hint next tile into cache
                __builtin_prefetch(Xb + (size_t)(bi * 64 + row) * Nn + kt + 64 + c4, 0, 0);
                __builtin_prefetch(Xb + (size_t)(bj * 64 + row) * Nn + kt + 64 + c4, 0, 0);
            }
        }
        __syncthreads();
#pragma unroll
        for (int k = 0; k < 64; k += 4) {
            // A 16x4 frag: lane L -> row L%16, K pair (k + (L/16)*2, +1)
            v2f a  = *(const v2f*)(&lds_i[(tr * 16 + rl) * 68 + k + half * 2]);
            // B 4x16 frag: lane L -> col L%16, same K-pair pattern (B = Xj rows)
            v2f bf = *(const v2f*)(&lds_j[(tc * 16 + rl) * 68 + k + half * 2]);
            acc = __builtin_amdgcn_wmma_f32_16x16x4_f32(false, a, false, bf,
                                                        (short)0, acc, false, false);
        }
        __syncthreads();
    }
    float* Mb = M + (size_t)b * Cc * Cc;
    const int r0 = bi * 64 + tr * 16;
    const int c0 = bj * 64 + tc * 16;
#pragma unroll
    for (int r = 0; r < 8; ++r) {
        const int m = r + half * 8;   // C/D layout: vgpr r -> rows r and r+8
        Mb[(size_t)(r0 + m) * Cc + c0 + rl] = acc[r];
    }
}

// ---------------------------------------------------------------------------
// Kernel 2: s[b][c] = sum_n xj[b,c,n]
// ---------------------------------------------------------------------------
__global__ __launch_bounds__(256) void k_rowsum(const float* __restrict__ xj,
                                                float* __restrict__ s) {
    const int b = blockIdx.y, c = blockIdx.x;
    const float* p = xj + ((size_t)b * Cc + c) * Nn;
    float partial = 0.f;
    for (int i = threadIdx.x; i < Nn; i += 256) partial += p[i];
    __shared__ float red[256];
    red[threadIdx.x] = partial;
    __syncthreads();
    for (int st = 128; st > 0; st >>= 1) {
        if (threadIdx.x < st) red[threadIdx.x] += red[threadIdx.x + st];
        __syncthreads();
    }
    if (threadIdx.x == 0) s[b * Cc + c] = red[0];
}

// ---------------------------------------------------------------------------
// Kernel 3: Gs_b = g_w @ s_b ; Ps_b = phi_w @ s_b
// ---------------------------------------------------------------------------
__global__ __launch_bounds__(256) void k_matvec(const float* __restrict__ g_w,
                                                const float* __restrict__ ph_w,
                                                const float* __restrict__ s,
                                                float* __restrict__ Gs,
                                                float* __restrict__ Ps) {
    const int b = blockIdx.y, which = blockIdx.x, i = threadIdx.x;
    const float* Mat = which ? ph_w : g_w;
    const float* sb  = s + b * Cc;
    float acc = 0.f;
    for (int k = 0; k < Cc; ++k) acc += Mat[i * Cc + k] * sb[k];
    (which ? Ps : Gs)[b * Cc + i] = acc;
}

// ---------------------------------------------------------------------------
// Kernel 4 (used twice): Out_b = Left @ Right_b   (Left shared, 256x256)
// ---------------------------------------------------------------------------
__global__ __launch_bounds__(256) void k_mm_left(const float* __restrict__ L,
                                                 const float* __restrict__ R,
                                                 float* __restrict__ O) {
    const int b = blockIdx.y, i = blockIdx.x, j = threadIdx.x;
    const float* Rb = R + (size_t)b * Cc * Cc;
    float acc = 0.f;
    for (int k = 0; k < Cc; ++k) acc += L[i * Cc + k] * Rb[k * Cc + j];
    O[(size_t)b * Cc * Cc + i * Cc + j] = acc;
}

// ---------------------------------------------------------------------------
// Kernel 5: A_b = T1_b @ P^T + (G s)bp^T + bg(P s)^T + N*bg*bp^T
// ---------------------------------------------------------------------------
__global__ __launch_bounds__(256) void k_mm_A(const float* __restrict__ T1,
                                              const float* __restrict__ P,
                                              const float* __restrict__ Gs,
                                              const float* __restrict__ Ps,
                                              const float* __restrict__ bg,
                                              const float* __restrict__ bp,
                                              float* __restrict__ Am) {
    const int b = blockIdx.y, i = blockIdx.x, j = threadIdx.x;
    const float* T1b = T1 + (size_t)b * Cc * Cc;
    float acc = 0.f;
    for (int k = 0; k < Cc; ++k) acc += T1b[i * Cc + k] * P[j * Cc + k];
    acc += Gs[b * Cc + i] * bp[j] + bg[i] * Ps[b * Cc + j]
         + (float)Nn * bg[i] * bp[j];
    Am[(size_t)b * Cc * Cc + i * Cc + j] = acc;
}

// ---------------------------------------------------------------------------
// Kernel 6: E_b = (T3_b @ theta_w)/N + I      (E' includes identity/residual)
// ---------------------------------------------------------------------------
__global__ __launch_bounds__(256) void k_mm_E(const float* __restrict__ T3,
                                              const float* __restrict__ Tw,
                                              float* __restrict__ Em) {
    const int b = blockIdx.y, i = blockIdx.x, j = threadIdx.x;
    const float* T3b = T3 + (size_t)b * Cc * Cc;
    float acc = 0.f;
    for (int k = 0; k < Cc; ++k) acc += T3b[i * Cc + k] * Tw[k * Cc + j];
    acc *= (1.0f / (float)Nn);
    if (i == j) acc += 1.0f;
    Em[(size_t)b * Cc * Cc + i * Cc + j] = acc;
}

// ---------------------------------------------------------------------------
// Kernel 7: e_b = (T3_b @ theta_b)/N + W_b
// ---------------------------------------------------------------------------
__global__ __launch_bounds__(256) void k_evec(const float* __restrict__ T3,
                                              const float* __restrict__ tb,
                                              const float* __restrict__ Wb,
                                              float* __restrict__ ev) {
    const int b = blockIdx.x, i = threadIdx.x;
    const float* T3b = T3 + (size_t)b * Cc * Cc;
    float acc = 0.f;
    for (int k = 0; k < Cc; ++k) acc += T3b[i * Cc + k] * tb[k];
    ev[b * Cc + i] = acc * (1.0f / (float)Nn) + Wb[i];
}

// ---------------------------------------------------------------------------
// Kernel 8: z_b = E'_b @ Xi_b + e_b  -> d_out  (pre-BN), fp32 WMMA
// Block = 512 threads (16 waves), tile 64 rows x 128 cols, Kt = 32.
// Xi tile staged transposed in LDS so B-frags are contiguous 8-byte reads.
// ---------------------------------------------------------------------------
__global__ __launch_bounds__(512) void k_zgemm(const float* __restrict__ Em,
                                               const float* __restrict__ xi,
                                               const float* __restrict__ ev,
                                               float* __restrict__ out) {
    const int b  = blockIdx.z;
    const int rb = blockIdx.y;   // 64-row block (0..3)
    const int nb = blockIdx.x;   // 128-col block (0..31)
    __shared__ float lds_e[64 * 36];    // E' rows x K, stride 36 (144 B rows)
    __shared__ float lds_x[128 * 36];   // Xi cols(n) x K (transposed)
    const int tid  = threadIdx.x;
    const int wv   = tid >> 5;
    const int lane = tid & 31;
    const int half = lane >> 4;
    const int rl   = lane & 15;
    const int tr   = wv >> 2;    // row tile 0..3
    const int tc0  = wv & 3;     // col tiles tc0 and tc0+4
    const float* Eb = Em + (size_t)b * Cc * Cc;
    const float* Xb = xi + (size_t)b * Cc * Nn;

    v8f acc0 = {}, acc1 = {};
    for (int kt = 0; kt < Cc; kt += 32) {
        {   // stage E': 64 rows x 32 K = 512 float4 slots, one per thread
            const int row = tid >> 3;
            const int c4  = (tid & 7) << 2;
            *(float4*)(&lds_e[row * 36 + c4]) =
                *(const float4*)(Eb + (size_t)(rb * 64 + row) * Cc + kt + c4);
        }
        // stage Xi transposed: 32 K-rows x 128 n-cols
        for (int slot = tid; slot < 32 * 32; slot += 512) {
            const int kr  = slot >> 5;
            const int nc4 = (slot & 31) << 2;
            const float4 v = *(const float4*)(Xb + (size_t)(kt + kr) * Nn + nb * 128 + nc4);
            lds_x[(nc4 + 0) * 36 + kr] = v.x;
            lds_x[(nc4 + 1) * 36 + kr] = v.y;
            lds_x[(nc4 + 2) * 36 + kr] = v.z;
            lds_x[(nc4 + 3) * 36 + kr] = v.w;
        }
        __syncthreads();
#pragma unroll
        for (int k = 0; k < 32; k += 4) {
            v2f a  = *(const v2f*)(&lds_e[(tr * 16 + rl) * 36 + k + half * 2]);
            v2f b0 = *(const v2f*)(&lds_x[(tc0 * 16 + rl) * 36 + k + half * 2]);
            v2f b1 = *(const v2f*)(&lds_x[((tc0 + 4) * 16 + rl) * 36 + k + half * 2]);
            acc0 = __builtin_amdgcn_wmma_f32_16x16x4_f32(false, a, false, b0,
                                                         (short)0, acc0, false, false);
            acc1 = __builtin_amdgcn_wmma_f32_16x16x4_f32(false, a, false, b1,
                                                         (short)0, acc1, false, false);
        }
        __syncthreads();
    }
    const int r0 = rb * 64 + tr * 16;
    float* outb = out + (size_t)b * Cc * Nn;
#pragma unroll
    for (int r = 0; r < 8; ++r) {
        const int m = r + half * 8;
        const float e = ev[b * Cc + r0 + m];
        const int n0 = nb * 128 + tc0 * 16 + rl;
        outb[(size_t)(r0 + m) * Nn + n0]      = acc0[r] + e;
        outb[(size_t)(r0 + m) * Nn + n0 + 64] = acc1[r] + e;
    }
}

// ---------------------------------------------------------------------------
// Kernel 9: per-channel sum / sumsq over (B, N)
// ---------------------------------------------------------------------------
__global__ __launch_bounds__(256) void k_bnstats(const float* __restrict__ z,
                                                 float* __restrict__ stats) {
    const int c = blockIdx.x;
    float s0 = 0.f, s1 = 0.f;
    for (int idx = threadIdx.x; idx < Bb * Nn; idx += 256) {
        const int b = idx >> 12;          // N = 4096 = 2^12
        const int n = idx & (Nn - 1);
        const float v = z[((size_t)b * Cc + c) * Nn + n];
        s0 += v;
        s1 += v * v;
    }
    __shared__ float r0[256], r1[256];
    r0[threadIdx.x] = s0;
    r1[threadIdx.x] = s1;
    __syncthreads();
    for (int st = 128; st > 0; st >>= 1) {
        if (threadIdx.x < st) {
            r0[threadIdx.x] += r0[threadIdx.x + st];
            r1[threadIdx.x] += r1[threadIdx.x + st];
        }
        __syncthreads();
    }
    if (threadIdx.x == 0) {
        stats[c]      = r0[0];
        stats[Cc + c] = r1[0];
    }
}

// ---------------------------------------------------------------------------
// Kernel 10: in-place BN normalize on d_out
// ---------------------------------------------------------------------------
__global__ __launch_bounds__(256) void k_bnnorm(float* __restrict__ z,
                                                const float* __restrict__ stats,
                                                const float* __restrict__ gamma,
                                                const float* __restrict__ beta) {
    const size_t idx = ((size_t)blockIdx.x << 8) + threadIdx.x;
    const int c = (int)((idx >> 12) & (Cc - 1));
    const float cnt  = (float)(Bb * Nn);
    const float mean = stats[c] / cnt;
    const float var  = stats[Cc + c] / cnt - mean * mean;
    const float inv  = rsqrtf(var + BN_EPS);
    const float v = z[idx];
    z[idx] = (v - mean) * inv * gamma[c] + beta[c];
}

// ---------------------------------------------------------------------------
extern "C" void kernel_launch(void* const* d_in, const int* in_sizes, int n_in,
                              void* d_out, int out_size, void* d_ws, size_t ws_size,
                              hipStream_t stream) {
    const float* xi   = (const float*)d_in[0];
    const float* xj   = (const float*)d_in[1];
    const float* g_w  = (const float*)d_in[2];
    const float* g_b  = (const float*)d_in[3];
    const float* th_w = (const float*)d_in[4];
    const float* th_b = (const float*)d_in[5];
    const float* ph_w = (const float*)d_in[6];
    const float* ph_b = (const float*)d_in[7];
    const float* W_w  = (const float*)d_in[8];
    const float* W_b  = (const float*)d_in[9];
    const float* gam  = (const float*)d_in[10];
    const float* bet  = (const float*)d_in[11];
    float* out = (float*)d_out;

    float* ws = (float*)d_ws;
    const size_t CC = (size_t)Cc * Cc;            // 65536
    float* M   = ws;                              // B*C*C
    float* T1  = M   + Bb * CC;
    float* Am  = T1  + Bb * CC;
    float* T3  = Am  + Bb * CC;
    float* Em  = T3  + Bb * CC;
    float* s   = Em  + Bb * CC;                   // B*C
    float* Gs  = s   + Bb * Cc;
    float* Ps  = Gs  + Bb * Cc;
    float* ev  = Ps  + Bb * Cc;
    float* st  = ev  + Bb * Cc;                   // 2*C

    // 1) M_b = Xj Xj^T   (WMMA fp32 SYRK)
    k_syrk<<<dim3(4, 4, Bb), 512, 0, stream>>>(xj, M);
    // 2) s_b = Xj @ 1
    k_rowsum<<<dim3(Cc, Bb), 256, 0, stream>>>(xj, s);
    // 3) Gs_b = G s_b ; Ps_b = P s_b
    k_matvec<<<dim3(2, Bb), 256, 0, stream>>>(g_w, ph_w, s, Gs, Ps);
    // 4) T1_b = G @ M_b
    k_mm_left<<<dim3(Cc, Bb), 256, 0, stream>>>(g_w, M, T1);
    // 5) A_b = T1 P^T + rank-1 bias terms
    k_mm_A<<<dim3(Cc, Bb), 256, 0, stream>>>(T1, ph_w, Gs, Ps, g_b, ph_b, Am);
    // 6) T3_b = W_w @ A_b
    k_mm_left<<<dim3(Cc, Bb), 256, 0, stream>>>(W_w, Am, T3);
    // 7) E'_b = (T3 theta_w)/N + I
    k_mm_E<<<dim3(Cc, Bb), 256, 0, stream>>>(T3, th_w, Em);
    // 8) e_b = (T3 theta_b)/N + W_b
    k_evec<<<Bb, 256, 0, stream>>>(T3, th_b, W_b, ev);
    // 9) z_b = E'_b Xi_b + e_b   (WMMA fp32 GEMM) -> d_out
    k_zgemm<<<dim3(Nn / 128, Cc / 64, Bb), 512, 0, stream>>>(Em, xi, ev, out);
    // 10) BN stats + normalize in place
    k_bnstats<<<Cc, 256, 0, stream>>>(out, st);
    k_bnnorm<<<(int)(((size_t)Bb * Cc * Nn) >> 8), 256, 0, stream>>>(out, st, gam, bet);
}